// mha_kvcache_38053410242950
// MI455X (gfx1250) — compile-verified
//
#include <hip/hip_runtime.h>
#include <hip/hip_bf16.h>

// ---------------------------------------------------------------------------
// MHA decode step with KV cache, f32 end-to-end, WMMA f32 16x16x4 matrix core.
// Memory-bound problem (~3.5 GB traffic -> ~150us floor @ 23.3 TB/s);
// f32 WMMA keeps ~25.8 GFLOP of GEMM well under the matrix ceiling
// (~0.6 PFLOPS f32 scaled from the FP8 WMMA rate).
// ---------------------------------------------------------------------------

typedef __attribute__((ext_vector_type(2))) float v2f;
typedef __attribute__((ext_vector_type(8))) float v8f;

#define BB   8
#define SS   16
#define HH   32
#define DH   128
#define HD   4096
#define PAST 4080
#define TOT  4096          // PAST + SS
#define MM   128           // BB*SS

// d_out layout (floats): out[524288] | attn[16777216] | K[134217728] | V[134217728]
#define OUT_N   524288
#define ATTN_N  16777216
#define KV_N    134217728

__device__ __forceinline__ v8f wmma_f32_16x16x4(v2f a, v2f b, v8f c) {
  // 8-arg pattern: (neg_a, A, neg_b, B, c_mod, C, reuse_a, reuse_b)
  return __builtin_amdgcn_wmma_f32_16x16x4_f32(false, a, false, b, (short)0, c,
                                               false, false);
}

// ---------------------------------------------------------------------------
// 1) Copy past KV into output cache (rows [0,4080) of each (b,h)).
//    2D grid: y = (b,h) slice, x covers the 130560 float4 of that slice.
//    No divides in the hot loop; pure global_load_b128 / global_store_b128.
// ---------------------------------------------------------------------------
__global__ __launch_bounds__(256) void kv_concat(const float* __restrict__ pk,
                                                 const float* __restrict__ pv,
                                                 float* __restrict__ Kout,
                                                 float* __restrict__ Vout) {
  const int per_bh_src = PAST * (DH / 4);                  // 130560 float4
  const int per_bh_dst = TOT * (DH / 4);                   // 131072 float4
  const int bh  = blockIdx.y;
  const int idx = blockIdx.x * 256 + threadIdx.x;          // [0, 130560)
  const float4* sk = (const float4*)pk + (size_t)bh * per_bh_src;
  const float4* sv = (const float4*)pv + (size_t)bh * per_bh_src;
  float4* dk = (float4*)Kout + (size_t)bh * per_bh_dst;
  float4* dv = (float4*)Vout + (size_t)bh * per_bh_dst;
  dk[idx] = sk[idx];
  dv[idx] = sv[idx];
}

// ---------------------------------------------------------------------------
// 2) OUT[m,n] = sum_k A[m,k] * W[n,k] + bias[n]     (x @ W.T + b)
//    Block = 256 thr (8 waves). Block tile: 16(M) x 128(N); wave owns 16x16.
//    K staged in 64-wide chunks through LDS (A: 4KB, W: 32KB), coalesced
//    global b128 loads + prefetch of the next chunk, ds_load_b64 fragment
//    reads, WMMA accumulate.
// ---------------------------------------------------------------------------
#define KT 64
__global__ __launch_bounds__(256) void gemm_xwT(const float* __restrict__ A,
                                                const float* __restrict__ W,
                                                const float* __restrict__ bias,
                                                float* __restrict__ OUT,
                                                int Kdim, int Ndim) {
  __shared__ float As[16 * KT];
  __shared__ float Ws[128 * KT];
  const int tid  = threadIdx.x;
  const int lane = tid & 31;
  const int wave = tid >> 5;
  const int lm   = lane & 15;          // M (or N) index inside fragment
  const int lk   = (lane >> 4) * 2;    // K sub-offset for upper half-wave
  const int m0   = blockIdx.x * 16;
  const int nblk = blockIdx.y * 128;
  const int n0   = nblk + wave * 16;

  v8f c = {};
  for (int k0 = 0; k0 < Kdim; k0 += KT) {
    { // A tile 16x64: thread t -> row t>>4, float4 at col (t&15)*4
      int r = tid >> 4, cc = (tid & 15) * 4;
      *(float4*)(As + r * KT + cc) =
          *(const float4*)(A + (size_t)(m0 + r) * Kdim + k0 + cc);
    }
    { // W tile 128x64: thread t -> row t>>1, 8 float4 starting at (t&1)*32
      int r = tid >> 1, cb = (tid & 1) * 32;
      const float* src = W + (size_t)(nblk + r) * Kdim + k0 + cb;
      float* dst = Ws + r * KT + cb;
#pragma unroll
      for (int i = 0; i < 8; ++i)
        *(float4*)(dst + i * 4) = *(const float4*)(src + i * 4);
      if (k0 + KT < Kdim)                    // global_prefetch_b8 next chunk
        __builtin_prefetch(src + KT, 0, 1);
    }
    __syncthreads();
#pragma unroll
    for (int k = 0; k < KT; k += 4) {
      v2f a = *(const v2f*)(As + lm * KT + k + lk);
      v2f b = *(const v2f*)(Ws + (wave * 16 + lm) * KT + k + lk);
      c = wmma_f32_16x16x4(a, b, c);
    }
    __syncthreads();
  }
  const float bvv = bias[n0 + lm];
#pragma unroll
  for (int r = 0; r < 8; ++r) {
    int mrow = m0 + r + 8 * (lane >> 4);       // C/D layout: VGPR r -> M=r,r+8
    OUT[(size_t)mrow * Ndim + n0 + lm] = c[r] + bvv;
  }
}

// ---------------------------------------------------------------------------
// 3) RoPE on Q and K; scatter new K/V rows into cache tail; Q -> (B,H,S,Dh).
// ---------------------------------------------------------------------------
__global__ __launch_bounds__(64) void rope_scatter(const float* __restrict__ Qb,
                                                   const float* __restrict__ Kb,
                                                   const float* __restrict__ Vb,
                                                   float* __restrict__ Qr,
                                                   float* __restrict__ Kout,
                                                   float* __restrict__ Vout) {
  const int bsh = blockIdx.x;             // B*S*H blocks
  const int h   = bsh % HH;
  const int bs  = bsh / HH;
  const int s   = bs % SS;
  const int b   = bs / SS;
  const int d   = threadIdx.x;            // 0..63 (rotation pair index)
  // inv_freq = 10000^(-d/64) = exp(-d * ln(10000)/64)
  const float inv = __expf(-(float)d * 0.14391165395f);
  const float ang = (float)(PAST + s) * inv;
  float sn, cs;
  __sincosf(ang, &sn, &cs);
  const size_t row = (size_t)(b * SS + s) * HD + (size_t)h * DH;
  const float q1 = Qb[row + d], q2 = Qb[row + d + 64];
  const float k1 = Kb[row + d], k2 = Kb[row + d + 64];
  const size_t qr = ((size_t)(b * HH + h) * SS + s) * DH;
  Qr[qr + d]      = q1 * cs - q2 * sn;
  Qr[qr + d + 64] = q2 * cs + q1 * sn;
  const size_t kv = ((size_t)(b * HH + h) * TOT + PAST + s) * DH;
  Kout[kv + d]      = k1 * cs - k2 * sn;
  Kout[kv + d + 64] = k2 * cs + k1 * sn;
  Vout[kv + d]      = Vb[row + d];
  Vout[kv + d + 64] = Vb[row + d + 64];
}

// ---------------------------------------------------------------------------
// 4) Attention per (b,h): scores = Q Kᵀ * scale (WMMA), softmax over 4096
//    with the whole 16x4096 score tile resident in LDS (256 KB of the WGP's
//    320 KB), then ctx = P V (WMMA, A-fragments via ds_load_b64 from LDS).
//    Q fragments (32 x v2f per wave) are preloaded once into registers.
// ---------------------------------------------------------------------------
__global__ __launch_bounds__(256) void attention(const float* __restrict__ Qr,
                                                 const float* __restrict__ Kf,
                                                 const float* __restrict__ Vf,
                                                 float* __restrict__ attn_g,
                                                 float* __restrict__ ctx) {
  extern __shared__ float smem[];
  float* sc      = smem;                  // 16 * 4096 scores
  float* red     = smem + 16 * TOT;       // 16 x 16 partials
  float* rowstat = red + 256;             // 16 row max / inv-sum

  const int bh   = blockIdx.x;
  const int h    = bh % HH;
  const int b    = bh / HH;
  const int tid  = threadIdx.x;
  const int lane = tid & 31;
  const int wave = tid >> 5;
  const int lm   = lane & 15;
  const int lk   = (lane >> 4) * 2;
  const float scale = 0.08838834764831845f;   // 128^-0.5

  const float* Qh = Qr + (size_t)bh * SS * DH;
  const float* Kh = Kf + (size_t)bh * TOT * DH;
  const float* Vh = Vf + (size_t)bh * TOT * DH;

  // ---- preload Q fragments: A[m][k] pairs for all 32 k-steps ----
  v2f qa[32];
#pragma unroll
  for (int i = 0; i < 32; ++i)
    qa[i] = *(const v2f*)(Qh + (size_t)lm * DH + i * 4 + lk);

  // ---- scores: wave w owns keys [w*512, w*512+512) in 32 tiles of 16 ----
  for (int kt = 0; kt < 32; ++kt) {
    const int key0 = wave * 512 + kt * 16;
    __builtin_prefetch(Kh + (size_t)(key0 + 16 + lm) * DH + lk, 0, 1);
    v8f c = {};
#pragma unroll
    for (int kd = 0; kd < DH; kd += 4) {
      v2f bf = *(const v2f*)(Kh + (size_t)(key0 + lm) * DH + kd + lk);
      c = wmma_f32_16x16x4(qa[kd >> 2], bf, c);
    }
#pragma unroll
    for (int r = 0; r < 8; ++r)
      sc[(r + 8 * (lane >> 4)) * TOT + key0 + lm] = c[r] * scale;
  }
  __syncthreads();

  // ---- softmax: 16 threads per row, 256 contiguous elems per thread ----
  const int row   = tid >> 4;
  const int chunk = tid & 15;
  float* p = sc + row * TOT + chunk * 256;
  float mx = -3.4e38f;
  for (int i = 0; i < 256; ++i) mx = fmaxf(mx, p[i]);
  red[row * 16 + chunk] = mx;
  __syncthreads();
  if (tid < 16) {
    float m = -3.4e38f;
    for (int i = 0; i < 16; ++i) m = fmaxf(m, red[tid * 16 + i]);
    rowstat[tid] = m;
  }
  __syncthreads();
  const float rmax = rowstat[row];
  float ssum = 0.f;
  for (int i = 0; i < 256; ++i) {
    float e = __expf(p[i] - rmax);
    p[i] = e;
    ssum += e;
  }
  red[row * 16 + chunk] = ssum;
  __syncthreads();
  if (tid < 16) {
    float s2 = 0.f;
    for (int i = 0; i < 16; ++i) s2 += red[tid * 16 + i];
    rowstat[tid] = 1.f / s2;
  }
  __syncthreads();
  const float rinv = rowstat[row];
  float* gout = attn_g + (size_t)bh * SS * TOT + row * TOT + chunk * 256;
  for (int i = 0; i < 256; i += 4) {
    float4 v = *(float4*)(p + i);
    v.x *= rinv; v.y *= rinv; v.z *= rinv; v.w *= rinv;
    *(float4*)(p + i)    = v;     // normalized probs stay in LDS for P@V
    *(float4*)(gout + i) = v;     // attn output (coalesced b128 stores)
  }
  __syncthreads();

  // ---- ctx = P @ V : wave w owns Dh columns [w*16, w*16+16) ----
  const int d0 = wave * 16;
  v8f acc = {};
#pragma unroll 4
  for (int kk = 0; kk < TOT; kk += 4) {
    v2f a = *(const v2f*)(sc + (size_t)lm * TOT + kk + lk);   // ds_load_b64
    v2f bf;
    bf.x = Vh[(size_t)(kk + lk) * DH + d0 + lm];
    bf.y = Vh[(size_t)(kk + lk + 1) * DH + d0 + lm];
    acc = wmma_f32_16x16x4(a, bf, acc);
  }
#pragma unroll
  for (int r = 0; r < 8; ++r) {
    int m = r + 8 * (lane >> 4);
    ctx[(size_t)(b * SS + m) * HD + (size_t)h * DH + d0 + lm] = acc[r];
  }
}

// ---------------------------------------------------------------------------
extern "C" void kernel_launch(void* const* d_in, const int* in_sizes, int n_in,
                              void* d_out, int out_size, void* d_ws,
                              size_t ws_size, hipStream_t stream) {
  const float* x  = (const float*)d_in[0];
  const float* pk = (const float*)d_in[1];
  const float* pv = (const float*)d_in[2];
  const float* Wq = (const float*)d_in[3];
  const float* bq = (const float*)d_in[4];
  const float* Wk = (const float*)d_in[5];
  const float* bk = (const float*)d_in[6];
  const float* Wv = (const float*)d_in[7];
  const float* bv = (const float*)d_in[8];
  const float* Wo = (const float*)d_in[9];
  const float* bo = (const float*)d_in[10];

  float* out  = (float*)d_out;
  float* attn = out + OUT_N;
  float* Kout = attn + ATTN_N;
  float* Vout = Kout + KV_N;

  float* ws   = (float*)d_ws;
  float* Qbuf = ws;                  // 128*4096
  float* Kbuf = ws + 524288;
  float* Vbuf = ws + 1048576;
  float* Qr   = ws + 1572864;        // (B,H,S,Dh)
  float* ctx  = ws + 2097152;        // (B*S, HD)

  // 510 blocks * 256 thr = 130560 float4 per (b,h) slice, 256 slices
  kv_concat<<<dim3(510, BB * HH), 256, 0, stream>>>(pk, pv, Kout, Vout);

  dim3 gg(MM / 16, HD / 128);        // (8, 32)
  gemm_xwT<<<gg, 256, 0, stream>>>(x, Wq, bq, Qbuf, HD, HD);
  gemm_xwT<<<gg, 256, 0, stream>>>(x, Wk, bk, Kbuf, HD, HD);
  gemm_xwT<<<gg, 256, 0, stream>>>(x, Wv, bv, Vbuf, HD, HD);

  rope_scatter<<<BB * SS * HH, 64, 0, stream>>>(Qbuf, Kbuf, Vbuf, Qr, Kout, Vout);

  const size_t smem_bytes = (16 * TOT + 256 + 16) * sizeof(float); // ~257 KB
  attention<<<BB * HH, 256, smem_bytes, stream>>>(Qr, Kout, Vout, attn, ctx);

  gemm_xwT<<<gg, 256, 0, stream>>>(ctx, Wo, bo, out, HD, HD);
}